// LightGCN_51874615001252
// MI455X (gfx1250) — compile-verified
//
#include <hip/hip_runtime.h>
#include <hip/hip_bf16.h>

#define EMB 128
#define NUM_PERSONS 100000
#define NUM_HOBBIES 50000
#define NTOT (NUM_PERSONS + NUM_HOBBIES)

typedef __attribute__((ext_vector_type(2))) float v2f;
typedef __attribute__((ext_vector_type(8))) float v8f;

// acc = cur = concat(person_emb, hobby_emb)
__global__ void lgcn_init_kernel(const float* __restrict__ pe,
                                 const float* __restrict__ he,
                                 float* __restrict__ acc,
                                 float* __restrict__ cur,
                                 int total) {
    int i = blockIdx.x * blockDim.x + threadIdx.x;
    if (i >= total) return;
    const int pcut = NUM_PERSONS * EMB;
    float v = (i < pcut) ? pe[i] : he[i - pcut];
    acc[i] = v;
    cur[i] = v;
}

__global__ void lgcn_zero_kernel(float* __restrict__ p, int total) {
    int i = blockIdx.x * blockDim.x + threadIdx.x;
    if (i < total) p[i] = 0.0f;
}

// One wave32 per edge. Lane l moves float4 at dim offset l*4.
// nxt[row] += val * cur[col]  via native global_atomic_add_f32.
__global__ void lgcn_spmm_kernel(const int* __restrict__ rows,
                                 const int* __restrict__ cols,
                                 const float* __restrict__ vals,
                                 const float* __restrict__ cur,
                                 float* __restrict__ nxt,
                                 int nnz) {
    int wave = (blockIdx.x * blockDim.x + threadIdx.x) >> 5;
    int lane = threadIdx.x & 31;
    if (wave >= nnz) return;

    int r = rows[wave];
    int c = cols[wave];
    float v = vals[wave];

    const float4* src = (const float4*)(cur + (size_t)c * EMB) + lane;
    float4 x = *src;

    float* dst = nxt + (size_t)r * EMB + lane * 4;
    unsafeAtomicAdd(dst + 0, v * x.x);
    unsafeAtomicAdd(dst + 1, v * x.y);
    unsafeAtomicAdd(dst + 2, v * x.z);
    unsafeAtomicAdd(dst + 3, v * x.w);
}

__global__ void lgcn_add_kernel(float* __restrict__ acc,
                                const float* __restrict__ nxt,
                                int total) {
    int i = blockIdx.x * blockDim.x + threadIdx.x;
    if (i < total) acc[i] += nxt[i];
}

// One wave32 per tile of 16 (person,hobby) pairs.
// C(16x16) = P(16x128) x H^T(128x16) via 32 x v_wmma_f32_16x16x4_f32,
// then out = diag(C) * (1/16)   [folds the /4 averaging on both operands].
//
// ISA layouts (§7.12.2), h = lane>=16 ? 1 : 0, sub = lane & 15:
//   A (16x4 f32):  lane sub = row M; VGPR0 holds K = 2h, VGPR1 holds K = 2h+1
//   B (4x16 f32):  lane sub = col N; VGPR0 holds K = 2h, VGPR1 holds K = 2h+1
//   C/D (16x16):   lanes 0-15: N=lane, VGPR r -> M=r; lanes 16-31: N=lane-16, VGPR r -> M=r+8
__global__ void lgcn_score_kernel(const int* __restrict__ pid,
                                  const int* __restrict__ hid,
                                  const float* __restrict__ acc,
                                  float* __restrict__ out,
                                  int batch) {
    int wave = (blockIdx.x * blockDim.x + threadIdx.x) >> 5;
    int lane = threadIdx.x & 31;
    int tile = wave;
    int base = tile * 16;
    if (base >= batch) return;

    int hgrp = lane >> 4;        // 0 or 1
    int sub  = lane & 15;

    int prow = pid[base + sub];
    int hrow = NUM_PERSONS + hid[base + sub];

    const float* pbase = acc + (size_t)prow * EMB + 2 * hgrp;
    const float* hbase = acc + (size_t)hrow * EMB + 2 * hgrp;

    v8f c = {};
#pragma unroll
    for (int kk = 0; kk < 32; ++kk) {
        v2f a, b;
        a.x = pbase[kk * 4 + 0];
        a.y = pbase[kk * 4 + 1];
        b.x = hbase[kk * 4 + 0];
        b.y = hbase[kk * 4 + 1];
        c = __builtin_amdgcn_wmma_f32_16x16x4_f32(
                /*neg_a=*/false, a, /*neg_b=*/false, b,
                /*c_mod=*/(short)0, c, /*reuse_a=*/false, /*reuse_b=*/false);
    }

    // Diagonal extraction: lanes 0-7 own diag entries 0-7 (element = lane),
    // lanes 24-31 own diag entries 8-15 (element = lane-24).
    bool lo = (lane < 8);
    bool hi = (lane >= 24);
    if (lo || hi) {
        int idx = lo ? lane : (lane - 24);
        float v = c[0];
        if (idx == 1) v = c[1];
        if (idx == 2) v = c[2];
        if (idx == 3) v = c[3];
        if (idx == 4) v = c[4];
        if (idx == 5) v = c[5];
        if (idx == 6) v = c[6];
        if (idx == 7) v = c[7];
        int o = lo ? (base + lane) : (base + lane - 16);
        out[o] = v * (1.0f / 16.0f);
    }
}

extern "C" void kernel_launch(void* const* d_in, const int* in_sizes, int n_in,
                              void* d_out, int out_size, void* d_ws, size_t ws_size,
                              hipStream_t stream) {
    const int*   person_ids = (const int*)d_in[0];
    const int*   hobby_ids  = (const int*)d_in[1];
    const int*   rows       = (const int*)d_in[2];
    const int*   cols       = (const int*)d_in[3];
    const float* vals       = (const float*)d_in[4];
    const float* person_emb = (const float*)d_in[5];
    const float* hobby_emb  = (const float*)d_in[6];
    float* out = (float*)d_out;

    const int nnz   = in_sizes[2];
    const int batch = in_sizes[0];
    const int total = NTOT * EMB;              // 19,200,000 floats
    const size_t NE = (size_t)NTOT * EMB;

    float* acc  = (float*)d_ws;                // 76.8 MB
    float* bufA = acc + NE;                    // cur
    float* bufB = bufA + NE;                   // nxt

    const int ethreads = 256;
    const int eblocks  = (total + ethreads - 1) / ethreads;

    lgcn_init_kernel<<<eblocks, ethreads, 0, stream>>>(person_emb, hobby_emb, acc, bufA, total);

    for (int layer = 0; layer < 3; ++layer) {
        lgcn_zero_kernel<<<eblocks, ethreads, 0, stream>>>(bufB, total);

        long long spmm_threads = (long long)nnz * 32;
        int spmm_blocks = (int)((spmm_threads + 255) / 256);
        lgcn_spmm_kernel<<<spmm_blocks, 256, 0, stream>>>(rows, cols, vals, bufA, bufB, nnz);

        lgcn_add_kernel<<<eblocks, ethreads, 0, stream>>>(acc, bufB, total);

        float* t = bufA; bufA = bufB; bufB = t;   // cur <- nxt (old cur re-zeroed next layer)
    }

    int tiles = (batch + 15) / 16;
    long long sthreads = (long long)tiles * 32;
    int sblocks = (int)((sthreads + 255) / 256);
    lgcn_score_kernel<<<sblocks, 256, 0, stream>>>(person_ids, hobby_ids, acc, out, batch);
}